// Model_84267258347802
// MI455X (gfx1250) — compile-verified
//
#include <hip/hip_runtime.h>
#include <math.h>

#define EPSF 1e-5f
#define BB 8
#define HH 128
#define WW2 128
#define HP 136
#define WP 136
#define PLANE (HP*WP)       /* 18496 */
#define WID 96
#define DIM 32
#define NMAT 136

typedef __attribute__((ext_vector_type(2))) float v2f;
typedef __attribute__((ext_vector_type(8))) float v8f;

__device__ __forceinline__ v8f wmma4(v2f a, v2f b, v8f c) {
  // V_WMMA_F32_16X16X4_F32 : D = A(16x4) * B(4x16) + C
  return __builtin_amdgcn_wmma_f32_16x16x4_f32(false, a, false, b, (short)0, c, false, false);
}
__device__ __forceinline__ float gelu_f(float x) {
  return 0.5f * x * (1.0f + erff(x * 0.7071067811865475f));
}

/* ------------------------------------------------------------------ */
/* Generic strided COMPLEX GEMM:  C[m][n] = sum_k A[m][k]*B[k][n] (+bias[m])
   A is row-major contiguous in k (acs==1). 2-D batch (d0,d1) strides.
   One 16x32 C strip per wave; K multiple of 4. Full tiles take a
   select-free loop; ragged tiles use load-then-cndmask (no exec branches). */
__global__ void cgemm_kernel(
    const float* __restrict__ Ar, const float* __restrict__ Ai,
    long long as0, long long as1, int ars,
    const float* __restrict__ Br, const float* __restrict__ Bi,
    long long bs0, long long bs1, int brs, int bcs,
    float* __restrict__ Cr, float* __restrict__ Ci,
    long long cs0, long long cs1, int crs, int ccs,
    const float* __restrict__ biasr, const float* __restrict__ biasi,
    int M, int N, int K, int d0, int d1)
{
  const int mt = (M + 15) >> 4, nt2 = (N + 31) >> 5;
  const long long total = (long long)d0 * d1 * mt * nt2;
  long long wid = (long long)blockIdx.x * (blockDim.x >> 5) + (threadIdx.x >> 5);
  if (wid >= total) return;
  const int lane = threadIdx.x & 31;
  const int l15 = lane & 15, hi = lane >> 4;

  long long t = wid;
  int ntile = (int)(t % nt2); t /= nt2;
  int mtile = (int)(t % mt); t /= mt;
  int i1 = (int)(t % d1);   int i0 = (int)(t / d1);

  const int arow = mtile * 16 + l15;
  const int col0 = ntile * 32 + l15;
  const int col1 = col0 + 16;
  const bool aok = arow < M;
  const bool b0k = col0 < N;
  const bool b1k = col1 < N;
  const int arowc = aok ? arow : 0;
  const int col0c = b0k ? col0 : 0;
  const int col1c = b1k ? col1 : 0;
  const bool fullTile = ((mtile * 16 + 16) <= M) && ((ntile * 32 + 32) <= N);

  const long long aoff = (long long)i0 * as0 + (long long)i1 * as1;
  const long long boff = (long long)i0 * bs0 + (long long)i1 * bs1;
  const float* par = Ar + aoff + (long long)arowc * ars + (hi * 2);
  const float* pai = Ai + aoff + (long long)arowc * ars + (hi * 2);
  const float* pbr0 = Br + boff + (long long)col0c * bcs + (long long)(hi * 2) * brs;
  const float* pbi0 = Bi + boff + (long long)col0c * bcs + (long long)(hi * 2) * brs;
  const float* pbr1 = Br + boff + (long long)col1c * bcs + (long long)(hi * 2) * brs;
  const float* pbi1 = Bi + boff + (long long)col1c * bcs + (long long)(hi * 2) * brs;
  const long long bstep = 4LL * brs;

  v8f P0 = {}, Q0 = {}, R0 = {}, S0 = {};
  v8f P1 = {}, Q1 = {}, R1 = {}, S1 = {};
  if (fullTile) {
    for (int kb = 0; kb < K; kb += 4) {
      __builtin_prefetch(pbr0 + 8 * bstep, 0, 3);
      __builtin_prefetch(pbi0 + 8 * bstep, 0, 3);
      const v2f ar = *(const v2f*)par;
      const v2f ai = *(const v2f*)pai;
      v2f br0, bi0, br1, bi1;
      br0.x = pbr0[0];  br0.y = pbr0[brs];
      bi0.x = pbi0[0];  bi0.y = pbi0[brs];
      br1.x = pbr1[0];  br1.y = pbr1[brs];
      bi1.x = pbi1[0];  bi1.y = pbi1[brs];
      P0 = wmma4(ar, br0, P0);  Q0 = wmma4(ai, bi0, Q0);
      R0 = wmma4(ar, bi0, R0);  S0 = wmma4(ai, br0, S0);
      P1 = wmma4(ar, br1, P1);  Q1 = wmma4(ai, bi1, Q1);
      R1 = wmma4(ar, bi1, R1);  S1 = wmma4(ai, br1, S1);
      par += 4; pai += 4;
      pbr0 += bstep; pbi0 += bstep; pbr1 += bstep; pbi1 += bstep;
    }
  } else {
    for (int kb = 0; kb < K; kb += 4) {
      const v2f arv = *(const v2f*)par;
      const v2f aiv = *(const v2f*)pai;
      const float r00 = pbr0[0], r01 = pbr0[brs];
      const float i00 = pbi0[0], i01 = pbi0[brs];
      const float r10 = pbr1[0], r11 = pbr1[brs];
      const float i10 = pbi1[0], i11 = pbi1[brs];
      v2f ar, ai, br0, bi0, br1, bi1;
      ar.x = aok ? arv.x : 0.f;  ar.y = aok ? arv.y : 0.f;
      ai.x = aok ? aiv.x : 0.f;  ai.y = aok ? aiv.y : 0.f;
      br0.x = b0k ? r00 : 0.f;   br0.y = b0k ? r01 : 0.f;
      bi0.x = b0k ? i00 : 0.f;   bi0.y = b0k ? i01 : 0.f;
      br1.x = b1k ? r10 : 0.f;   br1.y = b1k ? r11 : 0.f;
      bi1.x = b1k ? i10 : 0.f;   bi1.y = b1k ? i11 : 0.f;
      P0 = wmma4(ar, br0, P0);  Q0 = wmma4(ai, bi0, Q0);
      R0 = wmma4(ar, bi0, R0);  S0 = wmma4(ai, br0, S0);
      P1 = wmma4(ar, br1, P1);  Q1 = wmma4(ai, bi1, Q1);
      R1 = wmma4(ar, bi1, R1);  S1 = wmma4(ai, br1, S1);
      par += 4; pai += 4;
      pbr0 += bstep; pbi0 += bstep; pbr1 += bstep; pbi1 += bstep;
    }
  }
  float* crp = Cr + (long long)i0 * cs0 + (long long)i1 * cs1;
  float* cip = Ci + (long long)i0 * cs0 + (long long)i1 * cs1;
#pragma unroll
  for (int v = 0; v < 8; ++v) {
    const int row = mtile * 16 + hi * 8 + v;
    if (row < M) {
      const float bre = biasr ? biasr[row] : 0.f;
      const float bim = biasi ? biasi[row] : 0.f;
      const long long rb = (long long)row * crs;
      if (col0 < N) {
        const long long co = rb + (long long)col0 * ccs;
        crp[co] = P0[v] - Q0[v] + bre;
        cip[co] = R0[v] + S0[v] + bim;
      }
      if (col1 < N) {
        const long long co = rb + (long long)col1 * ccs;
        crp[co] = P1[v] - Q1[v] + bre;
        cip[co] = R1[v] + S1[v] + bim;
      }
    }
  }
}

/* Generic strided REAL GEMM, 2 N-subtiles per wave, optional pixel
   remap on B columns and GELU epilogue (fc1).                          */
__global__ void rgemm_kernel(
    const float* __restrict__ A, long long as0, int ars, int acs,
    const float* __restrict__ B, long long bs0, int brs, int bcs,
    int bremap, int brw, int brstride,
    float* __restrict__ C, long long cs0, int crs, int ccs,
    const float* __restrict__ bias, int act,
    int M, int N, int K, int d0)
{
  const int mt = (M + 15) >> 4, nt2 = (N + 31) >> 5;
  const long long total = (long long)d0 * mt * nt2;
  long long wid = (long long)blockIdx.x * (blockDim.x >> 5) + (threadIdx.x >> 5);
  if (wid >= total) return;
  const int lane = threadIdx.x & 31;
  const int l15 = lane & 15, hi = lane >> 4;

  long long t = wid;
  int ntile = (int)(t % nt2); t /= nt2;
  int mtile = (int)(t % mt); t /= mt;
  int i0 = (int)t;

  const int arow = mtile * 16 + l15;
  const int col0 = ntile * 32 + l15;
  const int col1 = col0 + 16;
  const bool aok = arow < M;
  const bool b0k = col0 < N;
  const bool b1k = col1 < N;
  const int arowc = aok ? arow : 0;
  const int col0c = b0k ? col0 : 0;
  const int col1c = b1k ? col1 : 0;
  const bool fullTile = ((mtile * 16 + 16) <= M) && ((ntile * 32 + 32) <= N);
  long long bb0, bb1;
  if (bremap) {
    bb0 = (long long)(col0c / brw) * brstride + (col0c % brw);
    bb1 = (long long)(col1c / brw) * brstride + (col1c % brw);
  } else {
    bb0 = (long long)col0c * bcs;
    bb1 = (long long)col1c * bcs;
  }
  const float* pa = A + (long long)i0 * as0 + (long long)arowc * ars + (long long)(hi * 2) * acs;
  const float* pb0 = B + (long long)i0 * bs0 + bb0 + (long long)(hi * 2) * brs;
  const float* pb1 = B + (long long)i0 * bs0 + bb1 + (long long)(hi * 2) * brs;
  const long long astep = 4LL * acs;
  const long long bstep = 4LL * brs;

  v8f P0 = {}, P1 = {};
  if (fullTile) {
    for (int kb = 0; kb < K; kb += 4) {
      __builtin_prefetch(pb0 + 8 * bstep, 0, 3);
      v2f a, b0, b1v;
      a.x = pa[0];    a.y = pa[acs];
      b0.x = pb0[0];  b0.y = pb0[brs];
      b1v.x = pb1[0]; b1v.y = pb1[brs];
      P0 = wmma4(a, b0, P0);
      P1 = wmma4(a, b1v, P1);
      pa += astep; pb0 += bstep; pb1 += bstep;
    }
  } else {
    for (int kb = 0; kb < K; kb += 4) {
      const float a0 = pa[0], a1 = pa[acs];
      const float c00 = pb0[0], c01 = pb0[brs];
      const float c10 = pb1[0], c11 = pb1[brs];
      v2f a, b0, b1v;
      a.x = aok ? a0 : 0.f;     a.y = aok ? a1 : 0.f;
      b0.x = b0k ? c00 : 0.f;   b0.y = b0k ? c01 : 0.f;
      b1v.x = b1k ? c10 : 0.f;  b1v.y = b1k ? c11 : 0.f;
      P0 = wmma4(a, b0, P0);
      P1 = wmma4(a, b1v, P1);
      pa += astep; pb0 += bstep; pb1 += bstep;
    }
  }
  float* cp = C + (long long)i0 * cs0;
#pragma unroll
  for (int v = 0; v < 8; ++v) {
    const int row = mtile * 16 + hi * 8 + v;
    if (row < M) {
      const float bs = bias ? bias[row] : 0.f;
      if (col0 < N) {
        float val = P0[v] + bs;
        if (act) val = gelu_f(val);
        cp[(long long)row * crs + (long long)col0 * ccs] = val;
      }
      if (col1 < N) {
        float val = P1[v] + bs;
        if (act) val = gelu_f(val);
        cp[(long long)row * crs + (long long)col1 * ccs] = val;
      }
    }
  }
}

/* ------------------------------------------------------------------ */
/* branch-free conv input load: clamp address, load, cndmask value     */
__device__ __forceinline__ float conv_ld(const float* __restrict__ xb,
                                         int ic, int yy, int xx) {
  const bool ok = (yy >= 0) & (yy < HP) & (xx >= 0) & (xx < WP);
  const int yc = yy < 0 ? 0 : (yy > HP - 1 ? HP - 1 : yy);
  const int xc = xx < 0 ? 0 : (xx > WP - 1 ? WP - 1 : xx);
  const float v = xb[(long long)ic * PLANE + yc * WP + xc];
  return ok ? v : 0.f;
}

/* Implicit-GEMM 3x3 conv (real), pad=1; 3 N-subtiles (48 px) per wave */
__global__ void conv3x3_r_kernel(const float* __restrict__ X,
                                 const float* __restrict__ Wt,
                                 const float* __restrict__ bias,
                                 float* __restrict__ Y, int C, int OC)
{
  const int octiles = OC >> 4;
  const long long total = (long long)BB * octiles * HP * 3;
  long long wid = (long long)blockIdx.x * (blockDim.x >> 5) + (threadIdx.x >> 5);
  if (wid >= total) return;
  const int lane = threadIdx.x & 31;
  const int l15 = lane & 15, hi = lane >> 4;
  long long t = wid;
  const int xt = (int)(t % 3); t /= 3;
  const int y  = (int)(t % HP); t /= HP;
  const int ot = (int)(t % octiles); const int b = (int)(t / octiles);
  const int K = C * 9;
  const float* wrow = Wt + (long long)(ot * 16 + l15) * K + hi * 2;
  const float* xb = X + (long long)b * C * PLANE;
  const int x0 = xt * 48;
  const int xp0 = x0 + l15, xp1 = xp0 + 16, xp2 = xp0 + 32;

  v8f P[3] = {};
  for (int kb = 0; kb < K; kb += 4) {
    const int ka = kb + hi * 2;
    const int ic0 = ka / 9;      const int r0 = ka - ic0 * 9;
    const int ky0 = r0 / 3;      const int kx0 = r0 - ky0 * 3;
    const int ic1 = (ka + 1) / 9; const int r1 = (ka + 1) - ic1 * 9;
    const int ky1 = r1 / 3;      const int kx1 = r1 - ky1 * 3;
    const int y0 = y + ky0 - 1, y1 = y + ky1 - 1;
    const v2f a = *(const v2f*)wrow;
    wrow += 4;
#pragma unroll
    for (int s = 0; s < 3; ++s) {
      const int xp = (s == 0) ? xp0 : (s == 1 ? xp1 : xp2);
      v2f bf;
      bf.x = conv_ld(xb, ic0, y0, xp + kx0 - 1);
      bf.y = conv_ld(xb, ic1, y1, xp + kx1 - 1);
      P[s] = wmma4(a, bf, P[s]);
    }
  }
#pragma unroll
  for (int s = 0; s < 3; ++s) {
    const int xx = (s == 0) ? xp0 : (s == 1 ? xp1 : xp2);
    if (xx < WP) {
#pragma unroll
      for (int v = 0; v < 8; ++v) {
        const int oc = ot * 16 + hi * 8 + v;
        Y[((long long)(b * OC + oc)) * PLANE + y * WP + xx] = P[s][v] + bias[oc];
      }
    }
  }
}

/* Implicit-GEMM 3x3 conv (complex), channel slices of a WID-ch tensor */
__global__ void conv3x3_c_kernel(const float* __restrict__ Xr,
                                 const float* __restrict__ Xi,
                                 int cin_tot, int cin_off, int C,
                                 const float* __restrict__ Wr,
                                 const float* __restrict__ Wi,
                                 const float* __restrict__ br,
                                 const float* __restrict__ bi,
                                 float* __restrict__ Yr, float* __restrict__ Yi,
                                 int cout_tot, int cout_off, int OC)
{
  const int octiles = OC >> 4;
  const long long total = (long long)BB * octiles * HP * 3;
  long long wid = (long long)blockIdx.x * (blockDim.x >> 5) + (threadIdx.x >> 5);
  if (wid >= total) return;
  const int lane = threadIdx.x & 31;
  const int l15 = lane & 15, hi = lane >> 4;
  long long t = wid;
  const int xt = (int)(t % 3); t /= 3;
  const int y  = (int)(t % HP); t /= HP;
  const int ot = (int)(t % octiles); const int b = (int)(t / octiles);
  const int K = C * 9;
  const float* wrr = Wr + (long long)(ot * 16 + l15) * K + hi * 2;
  const float* wir = Wi + (long long)(ot * 16 + l15) * K + hi * 2;
  const float* xrb = Xr + ((long long)b * cin_tot + cin_off) * PLANE;
  const float* xib = Xi + ((long long)b * cin_tot + cin_off) * PLANE;
  const int x0 = xt * 48;
  const int xp0 = x0 + l15, xp1 = xp0 + 16, xp2 = xp0 + 32;

  v8f P[3] = {}, Q[3] = {}, R[3] = {}, S[3] = {};
  for (int kb = 0; kb < K; kb += 4) {
    const int ka = kb + hi * 2;
    const int ic0 = ka / 9;       const int r0 = ka - ic0 * 9;
    const int ky0 = r0 / 3;       const int kx0 = r0 - ky0 * 3;
    const int ic1 = (ka + 1) / 9; const int r1 = (ka + 1) - ic1 * 9;
    const int ky1 = r1 / 3;       const int kx1 = r1 - ky1 * 3;
    const int y0 = y + ky0 - 1, y1 = y + ky1 - 1;
    const v2f ar = *(const v2f*)wrr;
    const v2f ai = *(const v2f*)wir;
    wrr += 4; wir += 4;
#pragma unroll
    for (int s = 0; s < 3; ++s) {
      const int xp = (s == 0) ? xp0 : (s == 1 ? xp1 : xp2);
      v2f brf, bif;
      brf.x = conv_ld(xrb, ic0, y0, xp + kx0 - 1);
      brf.y = conv_ld(xrb, ic1, y1, xp + kx1 - 1);
      bif.x = conv_ld(xib, ic0, y0, xp + kx0 - 1);
      bif.y = conv_ld(xib, ic1, y1, xp + kx1 - 1);
      P[s] = wmma4(ar, brf, P[s]);
      Q[s] = wmma4(ai, bif, Q[s]);
      R[s] = wmma4(ar, bif, R[s]);
      S[s] = wmma4(ai, brf, S[s]);
    }
  }
#pragma unroll
  for (int s = 0; s < 3; ++s) {
    const int xx = (s == 0) ? xp0 : (s == 1 ? xp1 : xp2);
    if (xx < WP) {
#pragma unroll
      for (int v = 0; v < 8; ++v) {
        const int oc = ot * 16 + hi * 8 + v;
        const long long o = ((long long)(b * cout_tot + cout_off + oc)) * PLANE + y * WP + xx;
        Yr[o] = P[s][v] - Q[s][v] + br[oc];
        Yi[o] = R[s][v] + S[s][v] + bi[oc];
      }
    }
  }
}

/* ------------------------------------------------------------------ */
/* fc0 + grid concat + transpose + zero-pad, directly to NCHW padded   */
__global__ void fc0_kernel(const float* __restrict__ x,
                           const float* __restrict__ w,
                           const float* __restrict__ b,
                           float* __restrict__ h)
{
  long long idx = (long long)blockIdx.x * blockDim.x + threadIdx.x;
  if (idx >= (long long)BB * PLANE) return;
  const int pix = (int)(idx % PLANE); const int bb = (int)(idx / PLANE);
  const int y = pix / WP, xx = pix % WP;
  if (y < HH && xx < WW2) {
    const long long xb = (((long long)bb * HH + y) * WW2 + xx) * 3;
    const float i0 = x[xb], i1 = x[xb + 1], i2 = x[xb + 2];
    const float gx = (float)y * (1.f / 127.f);
    const float gy = (float)xx * (1.f / 127.f);
    for (int c = 0; c < WID; ++c) {
      const float val = i0 * w[c] + i1 * w[WID + c] + i2 * w[2 * WID + c]
                      + gx * w[3 * WID + c] + gy * w[4 * WID + c] + b[c];
      h[((long long)(bb * WID + c)) * PLANE + pix] = val;
    }
  } else {
    for (int c = 0; c < WID; ++c)
      h[((long long)(bb * WID + c)) * PLANE + pix] = 0.f;
  }
}

/* per-channel mean/var over (B,HP,WP); one block per channel */
__global__ void bnstats_kernel(const float* __restrict__ X,
                               float* __restrict__ mean, float* __restrict__ var)
{
  const int c = blockIdx.x;
  __shared__ float s1[256], s2[256];
  float a = 0.f, q = 0.f;
  for (long long i = threadIdx.x; i < (long long)BB * PLANE; i += blockDim.x) {
    const int bb = (int)(i / PLANE); const int p = (int)(i % PLANE);
    const float v = X[((long long)(bb * WID + c)) * PLANE + p];
    a += v; q += v * v;
  }
  s1[threadIdx.x] = a; s2[threadIdx.x] = q;
  __syncthreads();
  for (int st = 128; st > 0; st >>= 1) {
    if ((int)threadIdx.x < st) { s1[threadIdx.x] += s1[threadIdx.x + st]; s2[threadIdx.x] += s2[threadIdx.x + st]; }
    __syncthreads();
  }
  if (threadIdx.x == 0) {
    const float inv = 1.f / (float)((long long)BB * PLANE);
    const float m = s1[0] * inv;
    mean[c] = m; var[c] = s2[0] * inv - m * m;
  }
}

__global__ void bn_gelu_kernel(const float* __restrict__ X,
                               const float* __restrict__ mean, const float* __restrict__ var,
                               const float* __restrict__ g, const float* __restrict__ beta,
                               float* __restrict__ Y)
{
  long long idx = (long long)blockIdx.x * blockDim.x + threadIdx.x;
  if (idx >= (long long)BB * WID * PLANE) return;
  const int c = (int)((idx / PLANE) % WID);
  const float v = (X[idx] - mean[c]) * rsqrtf(var[c] + EPSF) * g[c] + beta[c];
  Y[idx] = gelu_f(v);
}

/* complex BN stats: mr,mi,vrr,vii,vri per channel -> st[5*96] */
__global__ void cstats_kernel(const float* __restrict__ Xr, const float* __restrict__ Xi,
                              float* __restrict__ st)
{
  const int c = blockIdx.x;
  __shared__ float sh[5][256];
  float sr = 0.f, si = 0.f, srr = 0.f, sii = 0.f, sri = 0.f;
  for (long long i = threadIdx.x; i < (long long)BB * PLANE; i += blockDim.x) {
    const int bb = (int)(i / PLANE); const int p = (int)(i % PLANE);
    const long long o = ((long long)(bb * WID + c)) * PLANE + p;
    const float r = Xr[o], im = Xi[o];
    sr += r; si += im; srr += r * r; sii += im * im; sri += r * im;
  }
  sh[0][threadIdx.x] = sr;  sh[1][threadIdx.x] = si;
  sh[2][threadIdx.x] = srr; sh[3][threadIdx.x] = sii; sh[4][threadIdx.x] = sri;
  __syncthreads();
  for (int stp = 128; stp > 0; stp >>= 1) {
    if ((int)threadIdx.x < stp)
      for (int j = 0; j < 5; ++j) sh[j][threadIdx.x] += sh[j][threadIdx.x + stp];
    __syncthreads();
  }
  if (threadIdx.x == 0) {
    const float inv = 1.f / (float)((long long)BB * PLANE);
    const float mr = sh[0][0] * inv, mi = sh[1][0] * inv;
    st[c] = mr; st[96 + c] = mi;
    st[192 + c] = sh[2][0] * inv - mr * mr + EPSF;
    st[288 + c] = sh[3][0] * inv - mi * mi + EPSF;
    st[384 + c] = sh[4][0] * inv - mr * mi;
  }
}

__global__ void cbn_cgelu_kernel(const float* __restrict__ Xr, const float* __restrict__ Xi,
                                 const float* __restrict__ st,
                                 const float* __restrict__ grr, const float* __restrict__ gri,
                                 const float* __restrict__ gii,
                                 const float* __restrict__ betr, const float* __restrict__ beti,
                                 float* __restrict__ Yr, float* __restrict__ Yi)
{
  long long idx = (long long)blockIdx.x * blockDim.x + threadIdx.x;
  if (idx >= (long long)BB * WID * PLANE) return;
  const int c = (int)((idx / PLANE) % WID);
  const float mr = st[c], mi = st[96 + c];
  const float vrr = st[192 + c], vii = st[288 + c], vri = st[384 + c];
  const float s = sqrtf(vrr * vii - vri * vri);
  const float t = sqrtf(vrr + vii + 2.f * s);
  const float inv = 1.f / (s * t);
  const float rrr = (vii + s) * inv, rii = (vrr + s) * inv, rri = -vri * inv;
  const float zr = Xr[idx] - mr, zi = Xi[idx] - mi;
  const float yr = rrr * zr + rri * zi;
  const float yi = rri * zr + rii * zi;
  const float outr = grr[c] * yr + gri[c] * yi + betr[c];
  const float outi = gri[c] * yr + gii[c] * yi + beti[c];
  Yr[idx] = gelu_f(outr);
  Yi[idx] = gelu_f(outi);
}

__global__ void cgelu_kernel(float* __restrict__ Xr, float* __restrict__ Xi, long long n)
{
  long long i = (long long)blockIdx.x * blockDim.x + threadIdx.x;
  if (i < n) { Xr[i] = gelu_f(Xr[i]); Xi[i] = gelu_f(Xi[i]); }
}

/* mode-mixing einsum 'bixy,ioxy->boxy' on the 24x12 retained modes     */
__global__ void modemix_kernel(const float* __restrict__ A2r, const float* __restrict__ A2i,
                               const float* __restrict__ w1, const float* __restrict__ w2,
                               float* __restrict__ Mor, float* __restrict__ Moi)
{
  int idx = blockIdx.x * blockDim.x + threadIdx.x;
  if (idx >= BB * 32 * 32 * 16) return;
  const int cc = idx & 15;
  const int rr = (idx >> 4) & 31;
  const int o  = (idx >> 9) & 31;
  const int b  = idx >> 14;
  const long long out = (((long long)(b * 32 + o)) * 32 + rr) * 16 + cc;
  float accr = 0.f, acci = 0.f;
  if (rr < 24 && cc < 12) {
    const float* wsel = (rr < 12) ? w1 : w2;
    const int rr2 = (rr < 12) ? rr : rr - 12;
    for (int i = 0; i < 32; ++i) {
      const long long xi = (((long long)(b * 32 + i)) * 32 + rr) * 16 + cc;
      const float xr = A2r[xi], xim = A2i[xi];
      const long long wi = (((long long)i * 32 + o) * 12 + rr2) * 12 + cc;
      const float wr = wsel[2 * wi], wim = wsel[2 * wi + 1];
      accr += xr * wr - xim * wim;
      acci += xr * wim + xim * wr;
    }
  }
  Mor[out] = accr; Moi[out] = acci;
}

/* ------------------------------------------------------------------ */
/* DFRFT matrix construction: S matrix, parallel Jacobi eig, F build   */
__global__ void build_S_kernel(float* __restrict__ A, float* __restrict__ V)
{
  int idx = blockIdx.x * blockDim.x + threadIdx.x;
  if (idx >= NMAT * NMAT) return;
  const int m = idx / NMAT, n = idx % NMAT;
  float val = 0.f;
  if (m == n) val = 2.f * cosf(6.283185307179586f * (float)m / (float)NMAT);
  if (n == (m + 1) % NMAT) val += 1.f;
  if (n == (m - 1 + NMAT) % NMAT) val += 1.f;
  A[idx] = val;
  V[idx] = (m == n) ? 1.f : 0.f;
}

__global__ void jacobi_kernel(float* __restrict__ A, float* __restrict__ V)
{
  __shared__ float cs[68], sn[68];
  __shared__ int pp[68], qq[68];
  const int N = NMAT;
  for (int sweep = 0; sweep < 16; ++sweep) {
    for (int r = 0; r < N - 1; ++r) {
      if ((int)threadIdx.x < 68) {
        const int k = threadIdx.x;
        int p, q;
        if (k == 0) { p = N - 1; q = r % (N - 1); }
        else { p = (r + k) % (N - 1); q = (r - k + (N - 1)) % (N - 1); }
        if (p > q) { const int tt = p; p = q; q = tt; }
        const float app = A[p * N + p], aqq = A[q * N + q], apq = A[p * N + q];
        float c_ = 1.f, s_ = 0.f;
        if (fabsf(apq) > 1e-12f) {
          const float tau = (aqq - app) / (2.f * apq);
          const float tt = (tau >= 0.f) ? 1.f / (tau + sqrtf(1.f + tau * tau))
                                        : 1.f / (tau - sqrtf(1.f + tau * tau));
          c_ = rsqrtf(1.f + tt * tt);
          s_ = tt * c_;
        }
        cs[k] = c_; sn[k] = s_; pp[k] = p; qq[k] = q;
      }
      __syncthreads();
      for (int w = threadIdx.x; w < 68 * N; w += blockDim.x) {
        const int k = w / N, j = w - k * N;
        const int p = pp[k], q = qq[k];
        const float c_ = cs[k], s_ = sn[k];
        const float ap = A[p * N + j], aq = A[q * N + j];
        A[p * N + j] = c_ * ap - s_ * aq;
        A[q * N + j] = s_ * ap + c_ * aq;
      }
      __syncthreads();
      for (int w = threadIdx.x; w < 68 * N; w += blockDim.x) {
        const int k = w / N, j = w - k * N;
        const int p = pp[k], q = qq[k];
        const float c_ = cs[k], s_ = sn[k];
        const float ap = A[j * N + p], aq = A[j * N + q];
        A[j * N + p] = c_ * ap - s_ * aq;
        A[j * N + q] = s_ * ap + c_ * aq;
        const float vp = V[j * N + p], vq = V[j * N + q];
        V[j * N + p] = c_ * vp - s_ * vq;
        V[j * N + q] = s_ * vp + c_ * vq;
      }
      __syncthreads();
    }
  }
}

__global__ void sort_eig_kernel(const float* __restrict__ A, int* __restrict__ perm,
                                float* __restrict__ kv)
{
  if (threadIdx.x != 0 || blockIdx.x != 0) return;
  float ev[NMAT]; bool used[NMAT];
  for (int i = 0; i < NMAT; ++i) { ev[i] = A[i * NMAT + i]; used[i] = false; }
  for (int j = 0; j < NMAT; ++j) {
    int bi = 0; float bv = -3.0e38f;
    for (int i = 0; i < NMAT; ++i)
      if (!used[i] && ev[i] > bv) { bv = ev[i]; bi = i; }
    used[bi] = true; perm[j] = bi; kv[j] = (float)j;
  }
  kv[NMAT - 1] = (float)NMAT;
}

__global__ void build_F_kernel(const float* __restrict__ V, const int* __restrict__ perm,
                               const float* __restrict__ kv,
                               float* __restrict__ Fpr, float* __restrict__ Fpi,
                               float* __restrict__ Fnr, float* __restrict__ Fni)
{
  int idx = blockIdx.x * blockDim.x + threadIdx.x;
  if (idx >= 2 * NMAT * NMAT) return;
  const int mat = idx / (NMAT * NMAT);
  const int rem = idx % (NMAT * NMAT);
  const int m = rem / NMAT, n = rem % NMAT;
  const float alpha = mat ? -0.5f : 0.5f;
  float ar = 0.f, ai = 0.f;
  for (int j = 0; j < NMAT; ++j) {
    const int pj = perm[j];
    const float w = V[m * NMAT + pj] * V[n * NMAT + pj];
    const float ph = -1.5707963267948966f * alpha * kv[j];
    ar += w * cosf(ph);
    ai += w * sinf(ph);
  }
  if (mat == 0) { Fpr[rem] = ar; Fpi[rem] = ai; }
  else          { Fnr[rem] = ar; Fni[rem] = ai; }
}

/* truncated-DFT matrices for the spectral path */
__global__ void build_spec_kernel(float* WRr, float* WRi, float* WCr, float* WCi,
                                  float* ICr, float* ICi, float* IRr, float* IRi)
{
  int idx = blockIdx.x * blockDim.x + threadIdx.x;
  const float TP = 6.283185307179586f;
  const int nWR = 32 * NMAT, nWC = 16 * NMAT, nIC = 16 * NMAT, nIR = NMAT * 32;
  if (idx < nWR) {
    const int rr = idx / NMAT, y = idx % NMAT;
    float re = 0.f, im = 0.f;
    if (rr < 24) {
      const int f = (rr < 12) ? rr : 112 + rr;  /* 124 + (rr-12) */
      const float ang = -TP * (float)f * (float)y / (float)NMAT;
      re = cosf(ang); im = sinf(ang);
    }
    WRr[idx] = re; WRi[idx] = im;
  } else if (idx < nWR + nWC) {
    const int i2 = idx - nWR;
    const int cc = i2 / NMAT, x = i2 % NMAT;
    float re = 0.f, im = 0.f;
    if (cc < 12) {
      const float ang = -TP * (float)cc * (float)x / (float)NMAT;
      re = cosf(ang); im = sinf(ang);
    }
    WCr[i2] = re; WCi[i2] = im;
  } else if (idx < nWR + nWC + nIC) {
    const int i2 = idx - nWR - nWC;
    const int cc = i2 / NMAT, x = i2 % NMAT;
    float re = 0.f, im = 0.f;
    if (cc < 12) {
      const float ang = TP * (float)cc * (float)x / (float)NMAT;
      re = cosf(ang) / (float)NMAT; im = sinf(ang) / (float)NMAT;
    }
    ICr[i2] = re; ICi[i2] = im;
  } else if (idx < nWR + nWC + nIC + nIR) {
    const int i2 = idx - nWR - nWC - nIC;
    const int y = i2 / 32, rr = i2 % 32;
    float re = 0.f, im = 0.f;
    if (rr < 24) {
      const int f = (rr < 12) ? rr : 112 + rr;
      const float ang = TP * (float)f * (float)y / (float)NMAT;
      re = cosf(ang) / (float)NMAT; im = sinf(ang) / (float)NMAT;
    }
    IRr[i2] = re; IRi[i2] = im;
  }
}

__global__ void fc2_kernel(const float* __restrict__ fa, const float* __restrict__ w2,
                           const float* __restrict__ b2, float* __restrict__ out)
{
  int idx = blockIdx.x * blockDim.x + threadIdx.x;
  if (idx >= BB * HH * WW2) return;
  const int xx = idx % WW2;
  const int y = (idx / WW2) % HH;
  const int b = idx / (HH * WW2);
  float acc = b2[0];
  for (int j = 0; j < 128; ++j)
    acc += fa[((long long)(b * 128 + j)) * 16384 + y * 128 + xx] * w2[j];
  out[idx] = acc;
}

/* ------------------------------------------------------------------ */
static void cgemm(hipStream_t st,
                  const float* Ar, const float* Ai, long long as0, long long as1, int ars,
                  const float* Br, const float* Bi, long long bs0, long long bs1, int brs, int bcs,
                  float* Cr, float* Ci, long long cs0, long long cs1, int crs, int ccs,
                  const float* br, const float* bi,
                  int M, int N, int K, int d0, int d1)
{
  const long long waves = (long long)d0 * d1 * ((M + 15) / 16) * ((N + 31) / 32);
  const int blocks = (int)((waves + 7) / 8);
  cgemm_kernel<<<blocks, 256, 0, st>>>(Ar, Ai, as0, as1, ars,
                                       Br, Bi, bs0, bs1, brs, bcs,
                                       Cr, Ci, cs0, cs1, crs, ccs,
                                       br, bi, M, N, K, d0, d1);
}

extern "C" void kernel_launch(void* const* d_in, const int* in_sizes, int n_in,
                              void* d_out, int out_size, void* d_ws, size_t ws_size,
                              hipStream_t stream)
{
  (void)in_sizes; (void)n_in; (void)out_size; (void)ws_size;
  int ii = 0;
  const float* x     = (const float*)d_in[ii++];
  const float* fc0w  = (const float*)d_in[ii++];
  const float* fc0b  = (const float*)d_in[ii++];
  const float* g1    = (const float*)d_in[ii++];
  const float* b1    = (const float*)d_in[ii++];
  const float* cw1   = (const float*)d_in[ii++];
  const float* cc1   = (const float*)d_in[ii++];
  const float* g2    = (const float*)d_in[ii++];
  const float* b2p   = (const float*)d_in[ii++];
  const float* cw2   = (const float*)d_in[ii++];
  const float* cc2   = (const float*)d_in[ii++];
  const float* reswr = (const float*)d_in[ii++];
  const float* reswi = (const float*)d_in[ii++];
  const float* resbr = (const float*)d_in[ii++];
  const float* resbi = (const float*)d_in[ii++];
  const float* grr   = (const float*)d_in[ii++];
  const float* gri   = (const float*)d_in[ii++];
  const float* gii   = (const float*)d_in[ii++];
  const float* betr  = (const float*)d_in[ii++];
  const float* beti  = (const float*)d_in[ii++];
  const float* specw1[4]; const float* specw2[4];
  for (int i = 0; i < 4; ++i) { specw1[i] = (const float*)d_in[ii++]; specw2[i] = (const float*)d_in[ii++]; }
  const float* fcwr[4]; const float* fcwi[4]; const float* fcbr[4]; const float* fcbi[4];
  for (int i = 0; i < 4; ++i) {
    fcwr[i] = (const float*)d_in[ii++]; fcwi[i] = (const float*)d_in[ii++];
    fcbr[i] = (const float*)d_in[ii++]; fcbi[i] = (const float*)d_in[ii++];
  }
  const float* wcwr[4]; const float* wcwi[4]; const float* wcbr[4]; const float* wcbi[4];
  for (int i = 0; i < 4; ++i) {
    wcwr[i] = (const float*)d_in[ii++]; wcwi[i] = (const float*)d_in[ii++];
    wcbr[i] = (const float*)d_in[ii++]; wcbi[i] = (const float*)d_in[ii++];
  }
  const float* mixwr = (const float*)d_in[ii++];
  const float* mixwi = (const float*)d_in[ii++];
  const float* mixbr = (const float*)d_in[ii++];
  const float* mixbi = (const float*)d_in[ii++];
  const float* fc1w  = (const float*)d_in[ii++];
  const float* fc1b  = (const float*)d_in[ii++];
  const float* fc2w  = (const float*)d_in[ii++];
  const float* fc2b  = (const float*)d_in[ii++];
  float* out = (float*)d_out;

  /* workspace layout (floats) */
  const size_t TEN96 = (size_t)BB * WID * PLANE;   /* 14,204,928 */
  const size_t T32   = (size_t)256 * PLANE;        /* 4,734,976  */
  float* ws = (float*)d_ws;
  size_t off = 0;
  auto alloc = [&](size_t n) { float* p = ws + off; off += n; return p; };
  float* h   = alloc(TEN96);
  float* act = alloc(TEN96);
  float* z   = alloc(TEN96);
  float* cr  = alloc(TEN96);
  float* ci  = alloc(TEN96);
  float* nr  = alloc(TEN96);
  float* ni  = alloc(TEN96);
  float* t1r = alloc(T32);
  float* t1i = alloc(T32);
  float* t2r = alloc(T32);
  float* t2i = alloc(T32);
  float* A2r = alloc((size_t)256 * 512);
  float* A2i = alloc((size_t)256 * 512);
  float* Mor = alloc((size_t)256 * 512);
  float* Moi = alloc((size_t)256 * 512);
  float* Fpr = alloc(NMAT * NMAT); float* Fpi = alloc(NMAT * NMAT);
  float* Fnr = alloc(NMAT * NMAT); float* Fni = alloc(NMAT * NMAT);
  float* Smat = alloc(NMAT * NMAT);
  float* Vmat = alloc(NMAT * NMAT);
  int*   permI = (int*)alloc(NMAT);
  float* kv  = alloc(NMAT);
  float* WRr = alloc(32 * NMAT); float* WRi = alloc(32 * NMAT);
  float* WCr = alloc(16 * NMAT); float* WCi = alloc(16 * NMAT);
  float* ICr = alloc(16 * NMAT); float* ICi = alloc(16 * NMAT);
  float* IRr = alloc(NMAT * 32); float* IRi = alloc(NMAT * 32);
  float* bnm = alloc(WID); float* bnv = alloc(WID);
  float* cst = alloc(5 * WID);
  float* fc1a = t1r;  /* reuse t1..t2 region (18.9M floats >= 16.8M needed) */

  const long long CS = (long long)WID * PLANE;  /* per-batch ch stride */

  /* --- DFRFT + spectral matrix construction (one-time per launch) --- */
  build_S_kernel<<<(NMAT * NMAT + 255) / 256, 256, 0, stream>>>(Smat, Vmat);
  jacobi_kernel<<<1, 256, 0, stream>>>(Smat, Vmat);
  sort_eig_kernel<<<1, 1, 0, stream>>>(Smat, permI, kv);
  build_F_kernel<<<(2 * NMAT * NMAT + 255) / 256, 256, 0, stream>>>(Vmat, permI, kv, Fpr, Fpi, Fnr, Fni);
  build_spec_kernel<<<(13056 + 255) / 256, 256, 0, stream>>>(WRr, WRi, WCr, WCi, ICr, ICi, IRr, IRi);

  /* --- stem: fc0 -> pad -> conv block --- */
  fc0_kernel<<<((int)((long long)BB * PLANE + 255) / 256), 256, 0, stream>>>(x, fc0w, fc0b, h);
  const int ELT96 = (int)((TEN96 + 255) / 256);
  bnstats_kernel<<<WID, 256, 0, stream>>>(h, bnm, bnv);
  bn_gelu_kernel<<<ELT96, 256, 0, stream>>>(h, bnm, bnv, g1, b1, act);
  {
    const long long waves = (long long)BB * (WID / 16) * HP * 3;
    conv3x3_r_kernel<<<(int)((waves + 7) / 8), 256, 0, stream>>>(act, cw1, cc1, z, WID, WID);
  }
  bnstats_kernel<<<WID, 256, 0, stream>>>(z, bnm, bnv);
  bn_gelu_kernel<<<ELT96, 256, 0, stream>>>(z, bnm, bnv, g2, b2p, act);
  {
    const long long waves = (long long)BB * (WID / 16) * HP * 3;
    conv3x3_r_kernel<<<(int)((waves + 7) / 8), 256, 0, stream>>>(act, cw2, cc2, z, WID, WID);
  }

  /* --- res complex 1x1 conv on c = (h, z) -> (nr, ni) --- */
  cgemm(stream, reswr, reswi, 0, 0, WID,
        h, z, CS, 0, PLANE, 1,
        nr, ni, CS, 0, PLANE, 1,
        resbr, resbi, WID, PLANE, WID, BB, 1);
  cstats_kernel<<<WID, 256, 0, stream>>>(nr, ni, cst);
  cbn_cgelu_kernel<<<ELT96, 256, 0, stream>>>(nr, ni, cst, grr, gri, gii, betr, beti, cr, ci);

  /* --- 4 spectral/conv/dfrft iterations --- */
  for (int it = 0; it < 4; ++it) {
    /* x1: spectral path via truncated DFT, c[:, :32] -> n[:, :32] */
    cgemm(stream, WRr, WRi, 0, 0, NMAT,
          cr, ci, CS, PLANE, WP, 1,
          t1r, t1i, (long long)32 * 4352, 4352, NMAT, 1,
          nullptr, nullptr, 32, NMAT, NMAT, BB, 32);
    cgemm(stream, t1r, t1i, (long long)32 * 4352, 4352, NMAT,
          WCr, WCi, 0, 0, 1, NMAT,
          A2r, A2i, (long long)32 * 512, 512, 16, 1,
          nullptr, nullptr, 32, 16, NMAT, BB, 32);
    modemix_kernel<<<512, 256, 0, stream>>>(A2r, A2i, specw1[it], specw2[it], Mor, Moi);
    cgemm(stream, Mor, Moi, (long long)32 * 512, 512, 16,
          ICr, ICi, 0, 0, NMAT, 1,
          t1r, t1i, (long long)32 * 4352, 4352, NMAT, 1,
          nullptr, nullptr, 32, NMAT, 16, BB, 32);
    cgemm(stream, IRr, IRi, 0, 0, 32,
          t1r, t1i, (long long)32 * 4352, 4352, NMAT, 1,
          nr, ni, CS, PLANE, WP, 1,
          nullptr, nullptr, NMAT, NMAT, 32, BB, 32);

    /* x2: complex 3x3 conv on c[:, 32:64] -> n[:, 32:64] */
    {
      const long long waves = (long long)BB * (DIM / 16) * HP * 3;
      conv3x3_c_kernel<<<(int)((waves + 7) / 8), 256, 0, stream>>>(
          cr, ci, WID, DIM, DIM, wcwr[it], wcwi[it], wcbr[it], wcbi[it],
          nr, ni, WID, DIM, DIM);
    }

    /* x3: fractional DFT conv on c[:, 64:96] -> n[:, 64:96] */
    cgemm(stream, cr + 2 * DIM * PLANE, ci + 2 * DIM * PLANE, CS, PLANE, WP,
          Fpr, Fpi, 0, 0, 1, NMAT,
          t1r, t1i, (long long)DIM * PLANE, PLANE, WP, 1,
          nullptr, nullptr, NMAT, NMAT, NMAT, BB, DIM);
    cgemm(stream, Fpr, Fpi, 0, 0, NMAT,
          t1r, t1i, (long long)DIM * PLANE, PLANE, WP, 1,
          t2r, t2i, (long long)DIM * PLANE, PLANE, WP, 1,
          nullptr, nullptr, NMAT, NMAT, NMAT, BB, DIM);
    cgemm(stream, fcwr[it], fcwi[it], 0, 0, DIM,
          t2r, t2i, (long long)DIM * PLANE, 0, PLANE, 1,
          t1r, t1i, (long long)DIM * PLANE, 0, PLANE, 1,
          fcbr[it], fcbi[it], DIM, PLANE, DIM, BB, 1);
    cgemm(stream, Fnr, Fni, 0, 0, NMAT,
          t1r, t1i, (long long)DIM * PLANE, PLANE, WP, 1,
          t2r, t2i, (long long)DIM * PLANE, PLANE, WP, 1,
          nullptr, nullptr, NMAT, NMAT, NMAT, BB, DIM);
    cgemm(stream, t2r, t2i, (long long)DIM * PLANE, PLANE, WP,
          Fnr, Fni, 0, 0, 1, NMAT,
          nr + 2 * DIM * PLANE, ni + 2 * DIM * PLANE, CS, PLANE, WP, 1,
          nullptr, nullptr, NMAT, NMAT, NMAT, BB, DIM);

    /* mix 1x1 complex conv: c = Wmix * n + bias  (then cgelu if it<3) */
    cgemm(stream, mixwr, mixwi, 0, 0, WID,
          nr, ni, CS, 0, PLANE, 1,
          cr, ci, CS, 0, PLANE, 1,
          mixbr, mixbi, WID, PLANE, WID, BB, 1);
    if (it < 3)
      cgelu_kernel<<<ELT96, 256, 0, stream>>>(cr, ci, (long long)TEN96);
  }

  /* --- head: crop real part, fc1 (+gelu) via WMMA, fc2 --- */
  {
    const long long waves = (long long)BB * 8 * 512;
    rgemm_kernel<<<(int)((waves + 7) / 8), 256, 0, stream>>>(
        fc1w, 0, 1, 128,                  /* A[j][ch] = fc1w[ch*128+j]  */
        cr, CS, PLANE, 1, 1, 128, WP,     /* B with pixel remap 128->136 rows */
        fc1a, (long long)128 * 16384, 16384, 1,
        fc1b, 1, 128, 16384, WID, BB);
  }
  fc2_kernel<<<(BB * HH * WW2 + 255) / 256, 256, 0, stream>>>(fc1a, fc2w, fc2b, out);
}